// Conv2d_16320875725433
// MI455X (gfx1250) — compile-verified
//
#include <hip/hip_runtime.h>

typedef float v2f __attribute__((ext_vector_type(2)));
typedef float v8f __attribute__((ext_vector_type(8)));

#define BATCH   32
#define CIN     16
#define HH      224
#define WW      224
#define OW      222              // 224 - 2 (3x3, stride 1, no pad)
#define NWIN    (OW * OW)        // 49284 windows per batch
#define WINK    144              // C*KH*KW
#define OC      16
#define KSTEPS  (WINK / 4)       // 36 WMMA K-steps of 4
#define TILES_PER_B ((NWIN + 15) / 16)   // 3081 (last tile: 4 valid rows)
#define WAVES_PER_BLOCK 8
#define SW_STRIDE 146            // padded LDS stride for transposed weights (bank-conflict-free b64)

__global__ __launch_bounds__(256) void conv_im2col_wmma_f32(
    const float* __restrict__ x,
    const float* __restrict__ Wm,
    float* __restrict__ out)
{
    // LDS: transposed weights [OC][WINK] with padding, plus im2col offset table
    __shared__ float sWt[OC * SW_STRIDE];   // 16*146*4 = 9344 B
    __shared__ int   sOff[WINK];            // 144*4 B

    const int tid = threadIdx.x;

    // Stage weights transposed: sWt[n*146 + k] = Wm[k*16 + n]
    for (int i = tid; i < OC * WINK; i += 256) {
        int n = i / WINK;
        int k = i - n * WINK;
        sWt[n * SW_STRIDE + k] = Wm[k * OC + n];
    }
    // im2col offsets: off(k) = ch*H*W + kr*W + kc, order (channel, krow, kcol)
    for (int k = tid; k < WINK; k += 256) {
        int ch  = k / 9;
        int rem = k - ch * 9;
        int kr  = rem / 3;
        int kc  = rem - kr * 3;
        sOff[k] = ch * (HH * WW) + kr * WW + kc;
    }
    __syncthreads();

    const int lane = tid & 31;
    // tile id is uniform across the wave: force it into an SGPR so b/t/w0 and
    // all derived base pointers are scalar (SADDR-form loads/stores, s_cbranch).
    const int tile = __builtin_amdgcn_readfirstlane(blockIdx.x * WAVES_PER_BLOCK + (tid >> 5));
    if (tile >= BATCH * TILES_PER_B) return;

    const int b  = tile / TILES_PER_B;
    const int t  = tile - b * TILES_PER_B;
    const int w0 = t * 16;          // scalar

    const int m     = lane & 15;     // row within the 16-window tile (A) / oc column (B)
    const int khalf = lane >> 4;     // K pair selector per ISA A/B layout
    const int n     = m;             // output channel for B / D columns

    // Clamp gather row for the (scalar-known) partial last tile; stores handle bounds.
    int w = w0 + m;
    if (w > NWIN - 1) w = NWIN - 1;
    const int r = w / OW;
    const int c = w - r * OW;

    // Scalar 64-bit base + 32-bit per-lane element index -> GVS-mode global loads.
    const float* __restrict__ xbase = x + (size_t)b * (CIN * HH * WW);
    const int ibase = r * WW + c;    // per-lane, < 50176

    v8f acc = {0.f, 0.f, 0.f, 0.f, 0.f, 0.f, 0.f, 0.f};

    #pragma unroll 6
    for (int ks = 0; ks < KSTEPS; ++ks) {
        const int k0 = ks * 4 + khalf * 2;      // this lane's K pair: {k0, k0+1}

        // broadcast ds_load_b64: both im2col offsets
        const int2 offp = *(const int2*)&sOff[k0];
        // ds_load_b64: B pair (adjacent in transposed layout)
        const v2f  bp   = *(const v2f*)&sWt[n * SW_STRIDE + k0];

        v2f ap;
        ap.x = xbase[ibase + offp.x];
        ap.y = xbase[ibase + offp.y];

        // D = A(16x4) * B(4x16) + C   -> v_wmma_f32_16x16x4_f32
        acc = __builtin_amdgcn_wmma_f32_16x16x4_f32(
            /*neg_a=*/false, ap, /*neg_b=*/false, bp,
            /*c_mod=*/(short)0, acc, /*reuse_a=*/false, /*reuse_b=*/false);
    }

    // D layout: VGPR j -> M = j + 8*khalf, N = lane&15
    // Non-temporal stores: output is streamed once; keep x resident in the 192MB L2
    // (x is re-read ~9x by the im2col gather and fits entirely in L2).
    float* __restrict__ obase = out + ((size_t)b * NWIN + (size_t)w0) * OC;  // scalar base
    const int ostore = khalf * (8 * OC) + n;                                 // per-lane
    if (w0 + 16 <= NWIN) {
        // full tile: scalar-uniform fast path, unpredicated coalesced NT stores
        #pragma unroll
        for (int j = 0; j < 8; ++j)
            __builtin_nontemporal_store(acc[j], obase + ostore + j * OC);
    } else {
        const int mbase = khalf * 8;
        #pragma unroll
        for (int j = 0; j < 8; ++j) {
            if (w0 + mbase + j < NWIN)
                __builtin_nontemporal_store(acc[j], obase + ostore + j * OC);
        }
    }
}

extern "C" void kernel_launch(void* const* d_in, const int* in_sizes, int n_in,
                              void* d_out, int out_size, void* d_ws, size_t ws_size,
                              hipStream_t stream) {
    const float* x  = (const float*)d_in[0];   // [32,16,224,224] f32
    const float* Wm = (const float*)d_in[1];   // [144,16] f32
    float* out = (float*)d_out;                // [32,49284,16] f32

    const int total_tiles = BATCH * TILES_PER_B;              // 98592
    const int grid = (total_tiles + WAVES_PER_BLOCK - 1) / WAVES_PER_BLOCK;  // 12324
    conv_im2col_wmma_f32<<<grid, 256, 0, stream>>>(x, Wm, out);
}